// ganInstLayer_74594991996967
// MI455X (gfx1250) — compile-verified
//
#include <hip/hip_runtime.h>
#include <hip/hip_bf16.h>

typedef float v2f __attribute__((ext_vector_type(2)));
typedef float v8f __attribute__((ext_vector_type(8)));

#define NC   80
#define CH   256
#define HH   80
#define WW   80
#define NN   4096
#define M2   8192        // 2*N gathered rows
#define HWSZ (HH * WW)   // 6400
#define EPSC 1e-8f

// ---------------------------------------------------------------------------
// GEMM1: A = gathered feature columns (on the fly), B = w1^T, ReLU epilogue.
// One wave owns a 16x16 f32 tile; K=256 in steps of 4 via v_wmma_f32_16x16x4_f32.
// ---------------------------------------------------------------------------
__global__ __launch_bounds__(128) void gemm1_gather_kernel(
    const float* __restrict__ feature, const float* __restrict__ w1,
    const float* __restrict__ b1, const int* __restrict__ b_idx,
    const int* __restrict__ gj, const int* __restrict__ gi,
    const int* __restrict__ perm, float* __restrict__ H1)
{
  const int lane = threadIdx.x & 31;
  const int wave = threadIdx.x >> 5;
  const int tile = blockIdx.x * 4 + wave;   // 8192 tiles total
  const int mt = tile >> 4;                 // 0..511 (M tiles)
  const int nt = tile & 15;                 // 0..15  (N tiles)
  const int m0 = mt * 16, n0 = nt * 16;

  // Per-lane gathered-row base (lanes 16-31 mirror rows of lanes 0-15).
  const int r = m0 + (lane & 15);
  const int i = r & (NN - 1);
  int b = b_idx[i];
  if (r >= NN) b = perm[b];                 // domain-swapped half
  const float* fbase =
      feature + ((size_t)b * CH * HWSZ + (size_t)gj[i] * WW + gi[i]);

  const int n = n0 + (lane & 15);           // B column for this lane
  const int khalf = (lane < 16) ? 0 : 2;    // half-wave K split

  v8f acc = {};
  for (int k0 = 0; k0 < CH; k0 += 4) {
    const int klo = k0 + khalf;
    v2f a, bm;
    // A: 16x4 f32 tile, strided gather (channel stride = H*W), L2-resident.
    a.x = fbase[(size_t)klo * HWSZ];
    a.y = fbase[(size_t)(klo + 1) * HWSZ];
    // B: 4x16 tile of w1^T; w1 is [Cout,Cin] row-major -> contiguous float2.
    const float2 wv = *reinterpret_cast<const float2*>(w1 + (size_t)n * CH + klo);
    bm.x = wv.x; bm.y = wv.y;
    acc = __builtin_amdgcn_wmma_f32_16x16x4_f32(false, a, false, bm,
                                                (short)0, acc, false, false);
  }

  // Epilogue: +bias, ReLU, store. C/D layout: VGPR v -> row v (+8 for lanes>=16).
  const int col = n0 + (lane & 15);
  const float bias = b1[col];
  const int rowoff = (lane < 16) ? 0 : 8;
#pragma unroll
  for (int v = 0; v < 8; ++v) {
    const int row = m0 + v + rowoff;
    H1[(size_t)row * CH + col] = fmaxf(acc[v] + bias, 0.0f);
  }
}

// ---------------------------------------------------------------------------
// GEMM2: H1 [8192,256] x w2^T + b2 -> positive_sample rows 0..8191 (= fea2||fea2ba)
// ---------------------------------------------------------------------------
__global__ __launch_bounds__(128) void gemm2_kernel(
    const float* __restrict__ H1, const float* __restrict__ w2,
    const float* __restrict__ b2, float* __restrict__ ps)
{
  const int lane = threadIdx.x & 31;
  const int wave = threadIdx.x >> 5;
  const int tile = blockIdx.x * 4 + wave;
  const int mt = tile >> 4;
  const int nt = tile & 15;
  const int m0 = mt * 16, n0 = nt * 16;

  const int arow = m0 + (lane & 15);
  const int n = n0 + (lane & 15);
  const int khalf = (lane < 16) ? 0 : 2;

  v8f acc = {};
  for (int k0 = 0; k0 < CH; k0 += 4) {
    const int klo = k0 + khalf;
    v2f a, bm;
    const float2 av = *reinterpret_cast<const float2*>(H1 + (size_t)arow * CH + klo);
    a.x = av.x; a.y = av.y;
    const float2 wv = *reinterpret_cast<const float2*>(w2 + (size_t)n * CH + klo);
    bm.x = wv.x; bm.y = wv.y;
    acc = __builtin_amdgcn_wmma_f32_16x16x4_f32(false, a, false, bm,
                                                (short)0, acc, false, false);
  }

  const int col = n0 + (lane & 15);
  const float bias = b2[col];
  const int rowoff = (lane < 16) ? 0 : 8;
#pragma unroll
  for (int v = 0; v < 8; ++v) {
    const int row = m0 + v + rowoff;
    ps[(size_t)row * CH + col] = acc[v] + bias;
  }
}

// ---------------------------------------------------------------------------
// Cosine similarity: one wave32 per (fea2[i], fea2ba[i]) pair.
// ---------------------------------------------------------------------------
__global__ __launch_bounds__(256) void cosine_kernel(
    const float* __restrict__ fea, float* __restrict__ costy)
{
  const int lane = threadIdx.x & 31;
  const int wave = threadIdx.x >> 5;
  const int i = blockIdx.x * 8 + wave;      // 0..4095
  const float* x = fea + (size_t)i * CH;
  const float* y = fea + (size_t)(i + NN) * CH;

  float num = 0.f, na = 0.f, nb = 0.f;
#pragma unroll
  for (int c = lane; c < CH; c += 32) {
    const float xv = x[c], yv = y[c];
    num += xv * yv; na += xv * xv; nb += yv * yv;
  }
#pragma unroll
  for (int m = 16; m >= 1; m >>= 1) {
    num += __shfl_xor(num, m, 32);
    na  += __shfl_xor(na,  m, 32);
    nb  += __shfl_xor(nb,  m, 32);
  }
  if (lane == 0)
    costy[i] = num / fmaxf(sqrtf(na) * sqrtf(nb), EPSC);
}

// ---------------------------------------------------------------------------
// Per-class mean (deterministic, no float atomics): 1 block per class,
// thread = channel; tcls cached in LDS.
// ---------------------------------------------------------------------------
__global__ __launch_bounds__(256) void proto_kernel(
    const float* __restrict__ fea, const int* __restrict__ tcls,
    const float* __restrict__ proto, float* __restrict__ ps,
    float* __restrict__ present, float* __restrict__ proto_new)
{
  __shared__ int stcls[NN];
  for (int t = threadIdx.x; t < NN; t += 256) stcls[t] = tcls[t];
  __syncthreads();

  const int k = blockIdx.x;   // class
  const int c = threadIdx.x;  // channel
  float sum = 0.f;
  int cnt = 0;
  for (int r = 0; r < M2; ++r) {
    if (stcls[r & (NN - 1)] == k) { sum += fea[(size_t)r * CH + c]; ++cnt; }
  }
  const float val = (cnt > 0) ? (sum / (float)cnt) : proto[(size_t)k * CH + c];
  proto_new[(size_t)k * CH + c] = val;
  ps[(size_t)(M2 + k) * CH + c] = val;      // appended prototype rows
  if (c == 0) present[k] = (cnt > 0) ? 1.0f : 0.0f;
}

// ---------------------------------------------------------------------------
// positive_label fill.
// ---------------------------------------------------------------------------
__global__ __launch_bounds__(256) void labels_kernel(
    const int* __restrict__ tcls, float* __restrict__ plabel)
{
  const int idx = blockIdx.x * 256 + threadIdx.x;
  if (idx >= 2 * NN + NC) return;
  float v;
  if (idx < NN)          v = (float)tcls[idx];
  else if (idx < 2 * NN) v = (float)tcls[idx - NN];
  else                   v = (float)(idx - 2 * NN);
  plabel[idx] = v;
}

// ---------------------------------------------------------------------------
extern "C" void kernel_launch(void* const* d_in, const int* in_sizes, int n_in,
                              void* d_out, int out_size, void* d_ws, size_t ws_size,
                              hipStream_t stream) {
  const float* feature = (const float*)d_in[0];
  const float* w1      = (const float*)d_in[1];
  const float* b1      = (const float*)d_in[2];
  const float* w2      = (const float*)d_in[3];
  const float* b2      = (const float*)d_in[4];
  const float* proto   = (const float*)d_in[5];
  const int*   b_idx   = (const int*)d_in[6];
  const int*   gj      = (const int*)d_in[7];
  const int*   gi      = (const int*)d_in[8];
  const int*   tcls    = (const int*)d_in[9];
  const int*   perm    = (const int*)d_in[10];

  float* out     = (float*)d_out;
  float* costy   = out;                                  // [4096]
  float* ps      = out + NN;                             // [8272,256]
  float* plabel  = ps + (size_t)(M2 + NC) * CH;          // [8272]
  float* present = plabel + (M2 + NC);                   // [80]
  float* pnew    = present + NC;                         // [80,256]

  float* H1 = (float*)d_ws;                              // [8192,256] scratch

  // 8192 wave-tiles per GEMM, 4 waves per 128-thread block.
  gemm1_gather_kernel<<<2048, 128, 0, stream>>>(feature, w1, b1, b_idx, gj, gi,
                                                perm, H1);
  gemm2_kernel<<<2048, 128, 0, stream>>>(H1, w2, b2, ps);
  cosine_kernel<<<NN / 8, 256, 0, stream>>>(ps, costy);
  proto_kernel<<<NC, 256, 0, stream>>>(ps, tcls, proto, ps, present, pnew);
  labels_kernel<<<(2 * NN + NC + 255) / 256, 256, 0, stream>>>(tcls, plabel);
}